// MessageGraphConvolution_45019847197214
// MI455X (gfx1250) — compile-verified
//
#include <hip/hip_runtime.h>
#include <hip/hip_bf16.h>

#define FEAT 128
#define LDS_STRIDE 130   // 128 + 2 pad (keeps float2 alignment, spreads LDS banks)

typedef __attribute__((ext_vector_type(2))) float v2f;
typedef __attribute__((ext_vector_type(8))) float v8f;

// ---------------------------------------------------------------------------
// Kernel 1: zero the workspace (agg[N*128] ++ deg[N] are contiguous)
// ---------------------------------------------------------------------------
__global__ void mgc_zero_kernel(float* __restrict__ base, int total) {
    int i = blockIdx.x * blockDim.x + threadIdx.x;
    if (i < total) base[i] = 0.0f;
}

// ---------------------------------------------------------------------------
// Kernel 2: scatter-add. One wave (32 lanes) per edge; each lane moves a
// float4 of the 128-float message. Hardware f32 atomics -> L2 atomic units
// (agg is 25.6MB, L2-resident on the 192MB L2).
// ---------------------------------------------------------------------------
__global__ void mgc_scatter_kernel(const float* __restrict__ x,
                                   const int* __restrict__ src,
                                   const int* __restrict__ dst,
                                   float* __restrict__ agg,
                                   float* __restrict__ deg,
                                   int nEdges) {
    int lane = threadIdx.x & 31;
    int wave = blockIdx.x * (blockDim.x >> 5) + (threadIdx.x >> 5);
    if (wave >= nEdges) return;
    int s = src[wave];
    int d = dst[wave];
    const float4* xs = (const float4*)(x + (size_t)s * FEAT);
    float4 v = xs[lane];
    float* ap = agg + (size_t)d * FEAT + lane * 4;
    unsafeAtomicAdd(ap + 0, v.x);
    unsafeAtomicAdd(ap + 1, v.y);
    unsafeAtomicAdd(ap + 2, v.z);
    unsafeAtomicAdd(ap + 3, v.w);
    if (lane == 0) unsafeAtomicAdd(deg + d, 1.0f);
}

// ---------------------------------------------------------------------------
// Kernel 3: mean = agg / max(deg,1), in place, float4 per thread.
// ---------------------------------------------------------------------------
__global__ void mgc_normalize_kernel(float* __restrict__ agg,
                                     const float* __restrict__ deg,
                                     int nNodes) {
    int i = blockIdx.x * blockDim.x + threadIdx.x;   // over nNodes * 32 float4s
    int total = nNodes * (FEAT / 4);
    if (i >= total) return;
    int row = i >> 5;                                 // 32 float4 per row
    float dv = deg[row];
    float invd = (dv == 0.0f) ? 1.0f : (1.0f / dv);
    float4* a4 = (float4*)agg;
    float4 v = a4[i];
    v.x *= invd; v.y *= invd; v.z *= invd; v.w *= invd;
    a4[i] = v;
}

// ---------------------------------------------------------------------------
// Kernel 4: out = mean @ W^T + x @ B^T via V_WMMA_F32_16X16X4_F32.
// Block = 256 threads = 8 waves; block owns output rows [16*bid, 16*bid+16),
// wave w owns output cols [16*w, 16*w+16).  W and B (64KB each) are staged
// into LDS (dynamic, 130KB total) once per block; B-operands come from
// ds_load_b64, A-operands are coalesced global float2 loads.
//
// WMMA f32 16x16x4 operand layout (wave32):
//   A (16x4):  lane%16 = M row, {half=lane/16, vgpr} -> K = 2*half + vgpr
//   B (4x16):  lane%16 = N col, same K mapping
//   C/D:       vgpr v -> M = v + 8*half, lane%16 = N col
// ---------------------------------------------------------------------------
__global__ void __launch_bounds__(256)
mgc_gemm_kernel(const float* __restrict__ mean,
                const float* __restrict__ x,
                const float* __restrict__ W,
                const float* __restrict__ B,
                float* __restrict__ out) {
    extern __shared__ float smem[];
    float* sW = smem;
    float* sB = smem + FEAT * LDS_STRIDE;

    // Stage W and B: 8192 float2 each, coalesced, padded row stride in LDS.
    int t = threadIdx.x;
    for (int i = t; i < FEAT * FEAT / 2; i += 256) {
        int row = i >> 6;        // 64 float2 per 128-float row
        int k2  = i & 63;
        *(float2*)&sW[row * LDS_STRIDE + k2 * 2] = ((const float2*)W)[i];
        *(float2*)&sB[row * LDS_STRIDE + k2 * 2] = ((const float2*)B)[i];
    }
    __syncthreads();

    int lane   = t & 31;
    int waveId = t >> 5;                 // 0..7 -> column tile
    int l16    = lane & 15;
    int half   = lane >> 4;
    int row0   = blockIdx.x * 16;
    int col0   = waveId * 16;

    const float* mRow = mean + (size_t)(row0 + l16) * FEAT;  // A1 row (M = l16)
    const float* xRow = x    + (size_t)(row0 + l16) * FEAT;  // A2 row
    const float* wRow = sW + (col0 + l16) * LDS_STRIDE;      // B1[k][n] = W[n][k]
    const float* bRow = sB + (col0 + l16) * LDS_STRIDE;      // B2[k][n] = B[n][k]

    v8f acc = {0.f, 0.f, 0.f, 0.f, 0.f, 0.f, 0.f, 0.f};

#pragma unroll 4
    for (int k0 = 0; k0 < FEAT; k0 += 4) {
        int k = k0 + 2 * half;           // this lane's K pair for A and B
        v2f a1 = *(const v2f*)(mRow + k);
        v2f b1 = *(const v2f*)(wRow + k);
        acc = __builtin_amdgcn_wmma_f32_16x16x4_f32(false, a1, false, b1,
                                                    (short)0, acc, false, false);
        v2f a2 = *(const v2f*)(xRow + k);
        v2f b2 = *(const v2f*)(bRow + k);
        acc = __builtin_amdgcn_wmma_f32_16x16x4_f32(false, a2, false, b2,
                                                    (short)0, acc, false, false);
    }

    // D layout: vgpr v -> output row row0 + v + 8*half, col col0 + l16.
    float* o = out + (size_t)row0 * FEAT + col0 + l16;
#pragma unroll
    for (int v = 0; v < 8; v++) {
        o[(size_t)(v + 8 * half) * FEAT] = acc[v];
    }
}

// ---------------------------------------------------------------------------
extern "C" void kernel_launch(void* const* d_in, const int* in_sizes, int n_in,
                              void* d_out, int out_size, void* d_ws, size_t ws_size,
                              hipStream_t stream) {
    const float* x  = (const float*)d_in[0];
    const int*   ei = (const int*)d_in[1];     // [2, E]: sources then dests
    const float* W  = (const float*)d_in[2];
    const float* B  = (const float*)d_in[3];
    float*       out = (float*)d_out;

    const int N = in_sizes[0] / FEAT;          // 50000
    const int E = in_sizes[1] / 2;             // 1600000
    const int* src = ei;
    const int* dst = ei + E;

    float* agg = (float*)d_ws;                 // N*128 floats
    float* deg = agg + (size_t)N * FEAT;       // N floats

    // 1) zero workspace (every call: harness does not re-poison between runs)
    int totalZero = N * FEAT + N;
    mgc_zero_kernel<<<(totalZero + 255) / 256, 256, 0, stream>>>(agg, totalZero);

    // 2) scatter: one wave per edge, 8 waves per block
    int scatterBlocks = (E + 7) / 8;
    mgc_scatter_kernel<<<scatterBlocks, 256, 0, stream>>>(x, src, dst, agg, deg, E);

    // 3) normalize to mean (in place)
    int total4 = N * (FEAT / 4);
    mgc_normalize_kernel<<<(total4 + 255) / 256, 256, 0, stream>>>(agg, deg, N);

    // 4) WMMA GEMM: one block per 16-row stripe, 8 waves cover all 128 cols
    size_t shmem = (size_t)2 * FEAT * LDS_STRIDE * sizeof(float);  // ~130KB of 320KB WGP LDS
    mgc_gemm_kernel<<<N / 16, 256, shmem, stream>>>(agg, x, W, B, out);
}